// GNNnode_31069793419525
// MI455X (gfx1250) — compile-verified
//
#include <hip/hip_runtime.h>
#include <hip/hip_bf16.h>

// Shapes from the reference: batch=4, K+1=4, N=256, F=64 (f32 throughout).
#define NN   256
#define FF   64
#define KP1  4

typedef float v2f __attribute__((ext_vector_type(2)));
typedef float v8f __attribute__((ext_vector_type(8)));

// One block per (b, i). 256 threads = 8 wave32.
// Phase 1: thread t -> (jg = t>>4, fq = t&15); accumulate over 16 j's of its
//          stripe with float4 loads (fully coalesced 512B/wave), FMA with
//          A[b,i,j] broadcast from LDS.
// Phase 2: partials P[16][64] in LDS; waves 0..3 reduce the 16 stripes for a
//          16-wide f chunk using 4 chained V_WMMA_F32_16X16X4_F32 with an
//          all-ones A fragment (sum over K), add the hop-0 diagonal, store.
__global__ __launch_bounds__(256) void gnn_hopsum_kernel(
    const float* __restrict__ A,
    const float* __restrict__ X,
    float* __restrict__ Out)
{
    __shared__ float sA[NN];                                   // A row (1 KB)
    __shared__ __attribute__((aligned(16))) float sP[16 * FF]; // partials (4 KB)

    const int blk = blockIdx.x;   // b*NN + i
    const int b   = blk >> 8;
    const int i   = blk & (NN - 1);
    const int t   = threadIdx.x;

    // Stage the A row for this (b,i) into LDS (broadcast-read later).
    sA[t] = A[((size_t)b * NN + i) * NN + t];
    __syncthreads();

    const int fq = t & 15;   // which float4 within the 64-wide f row
    const int jg = t >> 4;   // j stripe 0..15 (16 j's each)

    // Row-major float4 views of X[b,k,i,:,:] for k = 1..3.
    const size_t rowElems = (size_t)NN * FF;  // elements per (b,k,i) slice
    const float4* __restrict__ r1 =
        (const float4*)(X + ((size_t)((b * KP1 + 1) * NN) + i) * rowElems);
    const float4* __restrict__ r2 =
        (const float4*)(X + ((size_t)((b * KP1 + 2) * NN) + i) * rowElems);
    const float4* __restrict__ r3 =
        (const float4*)(X + ((size_t)((b * KP1 + 3) * NN) + i) * rowElems);

    float4 acc = make_float4(0.f, 0.f, 0.f, 0.f);

#pragma unroll 8
    for (int jj = 0; jj < 16; ++jj) {
        const int j   = (jg << 4) + jj;
        const int idx = j * (FF / 4) + fq;

        if ((jj & 7) == 0 && jj + 8 < 16) {
            // Pull the next stripe chunk toward the caches (global_prefetch_b8).
            __builtin_prefetch(&r1[idx + 8 * (FF / 4)], 0, 0);
            __builtin_prefetch(&r2[idx + 8 * (FF / 4)], 0, 0);
            __builtin_prefetch(&r3[idx + 8 * (FF / 4)], 0, 0);
        }

        const float  a  = sA[j];
        const float4 p1 = r1[idx];
        const float4 p2 = r2[idx];
        const float4 p3 = r3[idx];

        acc.x = fmaf(a, p1.x + p2.x + p3.x, acc.x);
        acc.y = fmaf(a, p1.y + p2.y + p3.y, acc.y);
        acc.z = fmaf(a, p1.z + p2.z + p3.z, acc.z);
        acc.w = fmaf(a, p1.w + p2.w + p3.w, acc.w);
    }

    // P[jg][f] : partial sums per stripe.
    *(float4*)&sP[jg * FF + (fq << 2)] = acc;
    __syncthreads();

    // ---- WMMA reduction over the 16 stripes (waves 0..3, one f-chunk each) ----
    const int wid  = t >> 5;
    const int lane = t & 31;

    if (wid < 4) {                       // wave-uniform: EXEC all-ones inside
        const int cchunk = wid;                       // f chunk 0..3
        const int n      = (cchunk << 4) | (lane & 15); // f column this lane owns
        const int khalf  = (lane >> 4) << 1;          // 0 or 2: lane-half K offset

        v2f ones;
        ones.x = 1.0f;
        ones.y = 1.0f;

        v8f dacc = {};
#pragma unroll
        for (int s = 0; s < 4; ++s) {
            // B fragment: 4 consecutive stripe-partials (K=4), column n.
            // With an all-ones A, the sum is invariant to K-slot permutation,
            // so only the N = lane&15 mapping matters.
            const int k0 = (s << 2) + khalf;
            v2f bfrag;
            bfrag.x = sP[(k0 + 0) * FF + n];
            bfrag.y = sP[(k0 + 1) * FF + n];
            // D = ones(16x4) * P(4x16) + C   -> every row of D = stripe sum
            dacc = __builtin_amdgcn_wmma_f32_16x16x4_f32(
                /*neg_a=*/false, ones, /*neg_b=*/false, bfrag,
                /*c_mod=*/(short)0, dacc, /*reuse_a=*/false, /*reuse_b=*/false);
        }

        if ((lane & 31) < 16) {
            // Hop-0 contribution: only the diagonal element of X[b,0].
            const float diag =
                X[(((size_t)(b * KP1) * NN + i) * NN + i) * FF + n];
            Out[((size_t)b * NN + i) * FF + n] = dacc[0] + diag;
        }
    }
}

extern "C" void kernel_launch(void* const* d_in, const int* in_sizes, int n_in,
                              void* d_out, int out_size, void* d_ws, size_t ws_size,
                              hipStream_t stream) {
    const float* A = (const float*)d_in[0];
    const float* X = (const float*)d_in[1];
    float* Out     = (float*)d_out;

    const int batch = in_sizes[0] / (NN * NN);   // = 4
    dim3 grid(batch * NN), block(256);
    gnn_hopsum_kernel<<<grid, block, 0, stream>>>(A, X, Out);
}